// DendriteBranchLayer_63428077027567
// MI455X (gfx1250) — compile-verified
//
#include <hip/hip_runtime.h>
#include <math.h>

// ---------------- problem constants (match reference) ----------------
#define BATCH   4096
#define IN_E    4096
#define IN_I    4096
#define IN_BR   8192
#define DOUT    2048
#define KE      50
#define E_TO_I  50.0f
#define BN_EPS  1e-5f

#define TILE_B  8            // batch rows staged in LDS per workgroup (kernel 2)
#define ROWPAD  4097         // padded LDS row stride (floats) -> conflict-free gathers
#define NCH     32           // batch chunks for BN stats partials

typedef unsigned int u32x4 __attribute__((ext_vector_type(4)));
typedef int          i32x8 __attribute__((ext_vector_type(8)));
typedef int          i32x4 __attribute__((ext_vector_type(4)));
typedef float        v2f   __attribute__((ext_vector_type(2)));
typedef float        v8f   __attribute__((ext_vector_type(8)));

// ---------------- TDM: async 1-row (4096 f32) global->LDS tile load ----------------
// D# packing per CDNA5 ISA (2D tensor, tile = 1 x nelem).
// clang-23 toolchain: 6-arg builtin (g0, g1, g2, g3, g4, cpol).
__device__ __forceinline__ void tdm_load_row_f32(const float* gsrc, unsigned lds_byte, int nelem)
{
    unsigned long long ga = (unsigned long long)gsrc;
    u32x4 g0;
    g0.x = 1u;                                              // count=1 (valid), user mode
    g0.y = lds_byte;                                        // lds_addr[31:0]
    g0.z = (unsigned)(ga & 0xFFFFFFFFull);                  // global_addr[31:0]
    g0.w = (unsigned)((ga >> 32) & 0x1FFFFFFull)            // global_addr[56:32]
         | (2u << 30);                                      // type = 2 ("image")
    i32x8 g1;
    g1[0] = (2 << 16);                                      // wg_mask=0, data_size=2 (4B)
    g1[1] = (nelem & 0xFFFF) << 16;                         // tensor_dim0[15:0]
    g1[2] = ((nelem >> 16) & 0xFFFF) | (1 << 16);           // tensor_dim0[31:16], tensor_dim1=1
    g1[3] = (nelem & 0xFFFF) << 16;                         // tile_dim0 = nelem
    g1[4] = 1;                                              // tile_dim1 = 1, tile_dim2 = 0
    g1[5] = nelem;                                          // tensor_dim0_stride[31:0]
    g1[6] = 0;                                              // stride0[47:32], stride1[15:0]
    g1[7] = 0;
    i32x4 g2 = {0, 0, 0, 0};                                // 2D tensor: groups 2/3 unused
    i32x4 g3 = {0, 0, 0, 0};
    i32x8 g4 = {0, 0, 0, 0, 0, 0, 0, 0};
    __builtin_amdgcn_tensor_load_to_lds(g0, g1, g2, g3, g4, 0); // cpol = 0
}

// ---------------- Kernel 1: weight preprocessing ----------------
// One workgroup per output row d: top-50 of w_exc row, top-1 of w_inh row,
// and the 4 block-diagonal weights of w_block.
__global__ __launch_bounds__(256) void prep_kernel(
    const float* __restrict__ w_exc, const float* __restrict__ w_inh,
    const float* __restrict__ w_block,
    int* __restrict__ idxe, float* __restrict__ vale,
    int* __restrict__ idxi, float* __restrict__ vali_s,
    float* __restrict__ wbd)
{
    __shared__ float row[IN_E];
    __shared__ float rmax[256];
    __shared__ int   ridx[256];
    const int d = blockIdx.x;
    const int t = threadIdx.x;

    const float* we = w_exc + (size_t)d * IN_E;
    for (int i = t; i < IN_E; i += 256) row[i] = we[i];
    __syncthreads();

    for (int k = 0; k < KE; ++k) {
        float m = -3.402823466e+38f; int mi = 0;
        for (int i = t; i < IN_E; i += 256) {
            float v = row[i];
            if (v > m) { m = v; mi = i; }
        }
        rmax[t] = m; ridx[t] = mi;
        __syncthreads();
        for (int s = 128; s > 0; s >>= 1) {
            if (t < s && rmax[t + s] > rmax[t]) { rmax[t] = rmax[t + s]; ridx[t] = ridx[t + s]; }
            __syncthreads();
        }
        if (t == 0) {
            idxe[d * KE + k] = ridx[0];
            vale[d * KE + k] = rmax[0];
            row[ridx[0]] = -3.402823466e+38f;   // remove winner
        }
        __syncthreads();
    }

    // top-1 of w_inh row (store pre-scaled by -E_TO_I so main kernel is one FMA)
    const float* wi = w_inh + (size_t)d * IN_I;
    float m = -3.402823466e+38f; int mi = 0;
    for (int i = t; i < IN_I; i += 256) {
        float v = wi[i];
        if (v > m) { m = v; mi = i; }
    }
    rmax[t] = m; ridx[t] = mi;
    __syncthreads();
    for (int s = 128; s > 0; s >>= 1) {
        if (t < s && rmax[t + s] > rmax[t]) { rmax[t] = rmax[t + s]; ridx[t] = ridx[t + s]; }
        __syncthreads();
    }
    if (t == 0) { idxi[d] = ridx[0]; vali_s[d] = -E_TO_I * rmax[0]; }
    if (t < 4)  wbd[d * 4 + t] = w_block[(size_t)d * IN_BR + d * 4 + t];
}

// ---------------- Kernel 2: main sparse compute ----------------
// Workgroup = 8 waves; TDM stages 8 padded rows of x_e into LDS; each wave
// owns one batch row and sweeps all 2048 outputs (coalesced global I/O,
// random-but-padded LDS gathers for the 50 sparse FMAs).
__global__ __launch_bounds__(256) void dendrite_main(
    const float* __restrict__ xe, const float* __restrict__ xi,
    const float* __restrict__ xbr,
    const int* __restrict__ idxe, const float* __restrict__ vale,
    const int* __restrict__ idxi, const float* __restrict__ vali_s,
    const float* __restrict__ wbd, float* __restrict__ act)
{
    __shared__ float xerow[TILE_B * ROWPAD];   // ~131 KB
    const int b0 = blockIdx.x * TILE_B;

    if (threadIdx.x < 32) {                    // wave 0 issues the TDM DMAs
        for (int r = 0; r < TILE_B; ++r)
            tdm_load_row_f32(xe + (size_t)(b0 + r) * IN_E,
                             (unsigned)(r * ROWPAD * 4), IN_E);
        __builtin_amdgcn_s_wait_tensorcnt(0);
    }
    __syncthreads();

    const int bl   = threadIdx.x >> 5;         // wave id -> batch row in tile
    const int lane = threadIdx.x & 31;
    const int bg   = b0 + bl;

    const float* __restrict__ xrow  = &xerow[bl * ROWPAD];
    const float* __restrict__ xirow = xi  + (size_t)bg * IN_I;
    const float* __restrict__ xbrow = xbr + (size_t)bg * IN_BR;
    float*       __restrict__ arow  = act + (size_t)bg * DOUT;

    for (int d = lane; d < DOUT; d += 32) {
        const int*   ip = idxe + d * KE;
        const float* vp = vale + d * KE;
        float acc = 0.0f;
        #pragma unroll 10
        for (int k = 0; k < KE; ++k)
            acc = fmaf(xrow[ip[k]], vp[k], acc);

        // block-diagonal term: columns [4d, 4d+4), fully coalesced float4
        float4 xb = *(const float4*)(xbrow + d * 4);
        float4 wv = *(const float4*)(wbd + d * 4);
        acc = fmaf(xb.x, wv.x, acc);
        acc = fmaf(xb.y, wv.y, acc);
        acc = fmaf(xb.z, wv.z, acc);
        acc = fmaf(xb.w, wv.w, acc);

        // inhibition: single gather (x_i stays L2-resident), weight pre-scaled by -50
        acc = fmaf(xirow[idxi[d]], vali_s[d], acc);

        arow[d] = acc;
    }
}

// ---------------- Kernel 3: BN column stats via WMMA (ones x B = column sum) ----------
// One wave per (16-column slice, 128-batch chunk). A = ones(16x4) so
// D[m,n] = C[m,n] + sum_k B[k,n]: each WMMA folds 4 batch rows into the
// accumulator. Deterministic partials in workspace (no float atomics).
__global__ __launch_bounds__(32) void bn_stats_wmma(
    const float* __restrict__ act, float* __restrict__ sumP, float* __restrict__ sqP)
{
    const int d0   = blockIdx.x * 16;
    const int b0   = blockIdx.y * (BATCH / NCH);
    const int lane = threadIdx.x;
    const int col  = d0 + (lane & 15);
    const int half = lane >> 4;

    v8f cs = {};
    v8f cq = {};
    v2f a; a.x = 1.0f; a.y = 1.0f;            // all-ones 16x4 A matrix

    for (int b = b0; b < b0 + (BATCH / NCH); b += 4) {
        const int r = b + 2 * half;           // B layout: v0={K0,K2}, v1={K1,K3}
        v2f bv;
        bv.x = act[(size_t)r * DOUT + col];
        bv.y = act[(size_t)(r + 1) * DOUT + col];
        cs = __builtin_amdgcn_wmma_f32_16x16x4_f32(false, a, false, bv, (short)0, cs, false, false);
        v2f bq; bq.x = bv.x * bv.x; bq.y = bv.y * bv.y;
        cq = __builtin_amdgcn_wmma_f32_16x16x4_f32(false, a, false, bq, (short)0, cq, false, false);
    }
    if (lane < 16) {                          // D row M=0 lives in c[0], lanes 0-15
        sumP[blockIdx.y * DOUT + col] = cs[0];
        sqP [blockIdx.y * DOUT + col] = cq[0];
    }
}

// ---------------- Kernel 3b: fold stats + gamma/beta into scale/shift ----------------
__global__ __launch_bounds__(256) void bn_finalize(
    const float* __restrict__ sumP, const float* __restrict__ sqP,
    const float* __restrict__ gamma, const float* __restrict__ beta,
    float* __restrict__ scale, float* __restrict__ shift)
{
    const int d = blockIdx.x * 256 + threadIdx.x;
    float s = 0.0f, q = 0.0f;
    for (int c = 0; c < NCH; ++c) { s += sumP[c * DOUT + d]; q += sqP[c * DOUT + d]; }
    const float inv_n = 1.0f / (float)BATCH;
    const float mean  = s * inv_n;
    const float var   = q * inv_n - mean * mean;
    const float sc    = gamma[d] * rsqrtf(var + BN_EPS);
    scale[d] = sc;
    shift[d] = beta[d] - mean * sc;
}

// ---------------- Kernel 4: in-place BN apply + sigmoid ----------------
__global__ __launch_bounds__(256) void bn_apply_sigmoid(
    float* __restrict__ act, const float* __restrict__ scale, const float* __restrict__ shift)
{
    const size_t i = (size_t)blockIdx.x * 256 + threadIdx.x;
    const int d = (int)(i & (DOUT - 1));
    float v = fmaf(act[i], scale[d], shift[d]);
    act[i] = 1.0f / (1.0f + __expf(-v));
}

// ---------------- host launcher ----------------
extern "C" void kernel_launch(void* const* d_in, const int* in_sizes, int n_in,
                              void* d_out, int out_size, void* d_ws, size_t ws_size,
                              hipStream_t stream)
{
    const float* xe    = (const float*)d_in[0];
    const float* xi    = (const float*)d_in[1];
    const float* xbr   = (const float*)d_in[2];
    const float* w_exc = (const float*)d_in[3];
    const float* w_inh = (const float*)d_in[4];
    const float* w_blk = (const float*)d_in[5];
    const float* gamma = (const float*)d_in[6];
    const float* beta  = (const float*)d_in[7];

    // act (pre-BN) lives directly in d_out; kernel 4 rewrites it in place.
    float* act = (float*)d_out;

    // workspace layout (~1.4 MB, 16B-aligned offsets)
    char* ws = (char*)d_ws;
    int*   idxe   = (int*)  (ws + 0);                            // 2048*50 i32
    float* vale   = (float*)(ws + 409600);                       // 2048*50 f32
    int*   idxi   = (int*)  (ws + 819200);                       // 2048 i32
    float* vali_s = (float*)(ws + 827392);                       // 2048 f32 (pre * -50)
    float* wbd    = (float*)(ws + 835584);                       // 2048*4 f32
    float* sumP   = (float*)(ws + 868352);                       // NCH*2048 f32
    float* sqP    = (float*)(ws + 1130496);                      // NCH*2048 f32
    float* scale  = (float*)(ws + 1392640);                      // 2048 f32
    float* shift  = (float*)(ws + 1400832);                      // 2048 f32

    // 1) weight preprocessing: top-k selection + block-diag extraction
    prep_kernel<<<DOUT, 256, 0, stream>>>(w_exc, w_inh, w_blk,
                                          idxe, vale, idxi, vali_s, wbd);

    // 2) sparse dendrite compute (TDM-staged x_e tiles in LDS)
    dendrite_main<<<BATCH / TILE_B, 256, 0, stream>>>(
        xe, xi, xbr, idxe, vale, idxi, vali_s, wbd, act);

    // 3) BN batch statistics via WMMA column reduction
    dim3 sgrid(DOUT / 16, NCH);
    bn_stats_wmma<<<sgrid, 32, 0, stream>>>(act, sumP, sqP);

    // 3b) finalize scale/shift
    bn_finalize<<<DOUT / 256, 256, 0, stream>>>(sumP, sqP, gamma, beta, scale, shift);

    // 4) normalize + sigmoid in place
    bn_apply_sigmoid<<<(BATCH * DOUT) / 256, 256, 0, stream>>>(act, scale, shift);
}